// AttDeformF2F_25022479466675
// MI455X (gfx1250) — compile-verified
//
#include <hip/hip_runtime.h>
#include <hip/hip_bf16.h>

typedef float v2f __attribute__((ext_vector_type(2)));
typedef float v8f __attribute__((ext_vector_type(8)));

#define HH 32
#define WW 64
#define HWSZ 2048   // H*W, also attention sequence length L
#define NB 8        // batch
#define CC 128

static __device__ inline v8f wmma4(v2f a, v2f b, v8f c) {
  // D = A(16x4,f32) * B(4x16,f32) + C(16x16,f32)
  return __builtin_amdgcn_wmma_f32_16x16x4_f32(false, a, false, b, (short)0, c,
                                               false, false);
}

// ---------------------------------------------------------------------------
// Channel-major GEMM:  out[n][o][l] = act( sum_k in[n][k][l] * w[o*K+k] + b[o] )
// One wave per 32(M) x 32(N) register tile (4 accumulators, A & B shared).
// grid = (N*L/32, O/32), block = 32.
// ---------------------------------------------------------------------------
__global__ __launch_bounds__(32) void k_gemm_cm(const float* __restrict__ in,
                                                const float* __restrict__ w,
                                                const float* __restrict__ bias,
                                                float* __restrict__ out,
                                                int K, int O, int relu) {
  const int L = HWSZ;
  int lane = threadIdx.x & 31;
  int lr = lane & 15;
  int hi = lane >> 4;          // 0: K-sub 0,1   1: K-sub 2,3 (A & B operands)
  int kofs = hi << 1;
  int mt = blockIdx.x;
  int n = mt >> 6;             // 64 M-tiles (of 32 rows) per image
  int lbase = (mt & 63) << 5;
  int nc0 = (blockIdx.y << 5) + lr;
  int nc1 = nc0 + 16;
  const float* ain0 = in + (size_t)n * K * L + lbase + lr;  // A[m][k]=ain[k*L]
  const float* ain1 = ain0 + 16;
  const float* w0 = w + (size_t)nc0 * K;
  const float* w1 = w + (size_t)nc1 * K;
  v8f c00 = {}, c01 = {}, c10 = {}, c11 = {};
  for (int k = 0; k < K; k += 4) {
    size_t ko = (size_t)(k + kofs) * L;
    v2f a0, a1, b0, b1;
    a0.x = ain0[ko];      a0.y = ain0[ko + L];
    a1.x = ain1[ko];      a1.y = ain1[ko + L];
    b0.x = w0[k + kofs];  b0.y = w0[k + kofs + 1];
    b1.x = w1[k + kofs];  b1.y = w1[k + kofs + 1];
    c00 = wmma4(a0, b0, c00);
    c01 = wmma4(a0, b1, c01);
    c10 = wmma4(a1, b0, c10);
    c11 = wmma4(a1, b1, c11);
  }
  float bb0 = bias[nc0], bb1 = bias[nc1];
  int rowb = hi << 3;
#pragma unroll
  for (int v = 0; v < 8; v++) {
    float t00 = c00[v] + bb0, t01 = c01[v] + bb1;
    float t10 = c10[v] + bb0, t11 = c11[v] + bb1;
    if (relu) {
      t00 = fmaxf(t00, 0.f); t01 = fmaxf(t01, 0.f);
      t10 = fmaxf(t10, 0.f); t11 = fmaxf(t11, 0.f);
    }
    c00[v] = t00; c01[v] = t01; c10[v] = t10; c11[v] = t11;
  }
  float* o0 = out + ((size_t)n * O + nc0) * L + lbase + rowb;
  float* o1 = out + ((size_t)n * O + nc1) * L + lbase + rowb;
  *(float4*)(o0)          = make_float4(c00[0], c00[1], c00[2], c00[3]);
  *(float4*)(o0 + 4)      = make_float4(c00[4], c00[5], c00[6], c00[7]);
  *(float4*)(o0 + 16)     = make_float4(c10[0], c10[1], c10[2], c10[3]);
  *(float4*)(o0 + 20)     = make_float4(c10[4], c10[5], c10[6], c10[7]);
  *(float4*)(o1)          = make_float4(c01[0], c01[1], c01[2], c01[3]);
  *(float4*)(o1 + 4)      = make_float4(c01[4], c01[5], c01[6], c01[7]);
  *(float4*)(o1 + 16)     = make_float4(c11[0], c11[1], c11[2], c11[3]);
  *(float4*)(o1 + 20)     = make_float4(c11[4], c11[5], c11[6], c11[7]);
}

// ---------------------------------------------------------------------------
// 3x3 conv (pad 1) as implicit GEMM, K = IC*9, ordered k = kk*IC + c.
// 32(M) x 32(N) register tile per wave, tap geometry hoisted per M-subtile.
// grid = (N*HW/32, ceil(OC/32)), block = 32.
// ---------------------------------------------------------------------------
__global__ __launch_bounds__(32) void k_conv3x3(const float* __restrict__ in,
                                                const float* __restrict__ w,
                                                const float* __restrict__ bias,
                                                float* __restrict__ out,
                                                int IC, int OC, int relu) {
  int lane = threadIdx.x & 31;
  int lr = lane & 15;
  int hi = lane >> 4;
  int kofs = hi << 1;
  int mt = blockIdx.x;
  int n = mt >> 6;
  int lbase = (mt & 63) << 5;
  int y = lbase >> 6;                  // 32-aligned tile stays in one row
  int x0 = (lbase & 63) + lr;
  int x1 = x0 + 16;
  int ncol0 = (blockIdx.y << 5) + lr;
  int ncol1 = ncol0 + 16;
  int ncc0 = ncol0 < OC ? ncol0 : (OC - 1);
  int ncc1 = ncol1 < OC ? ncol1 : (OC - 1);
  const float* nbase = in + (size_t)n * IC * HWSZ;
  v8f c00 = {}, c01 = {}, c10 = {}, c11 = {};
  for (int kk = 0; kk < 9; kk++) {
    int sy = y + (kk / 3) - 1;
    int vy = (sy >= 0) & (sy < HH);
    int cy = sy < 0 ? 0 : (sy >= HH ? HH - 1 : sy);
    int sx0 = x0 + (kk % 3) - 1;
    int sx1 = x1 + (kk % 3) - 1;
    float vm0 = (float)(vy & (sx0 >= 0) & (sx0 < WW));
    float vm1 = (float)(vy & (sx1 >= 0) & (sx1 < WW));
    int cx0 = sx0 < 0 ? 0 : (sx0 >= WW ? WW - 1 : sx0);
    int cx1 = sx1 < 0 ? 0 : (sx1 >= WW ? WW - 1 : sx1);
    const float* ap0 = nbase + cy * WW + cx0;           // + c*HWSZ per channel
    const float* ap1 = nbase + cy * WW + cx1;
    const float* bp0 = w + (size_t)ncc0 * IC * 9 + kk;  // + c*9 per channel
    const float* bp1 = w + (size_t)ncc1 * IC * 9 + kk;
    for (int c = 0; c < IC; c += 4) {
      int cc = c + kofs;
      v2f a0, a1, b0, b1;
      a0.x = vm0 * ap0[(size_t)cc * HWSZ];
      a0.y = vm0 * ap0[(size_t)(cc + 1) * HWSZ];
      a1.x = vm1 * ap1[(size_t)cc * HWSZ];
      a1.y = vm1 * ap1[(size_t)(cc + 1) * HWSZ];
      b0.x = bp0[(size_t)cc * 9];
      b0.y = bp0[(size_t)(cc + 1) * 9];
      b1.x = bp1[(size_t)cc * 9];
      b1.y = bp1[(size_t)(cc + 1) * 9];
      c00 = wmma4(a0, b0, c00);
      c01 = wmma4(a0, b1, c01);
      c10 = wmma4(a1, b0, c10);
      c11 = wmma4(a1, b1, c11);
    }
  }
  float bb0 = bias[ncc0], bb1 = bias[ncc1];
  int rowb = hi << 3;
#pragma unroll
  for (int v = 0; v < 8; v++) {
    float t00 = c00[v] + bb0, t01 = c01[v] + bb1;
    float t10 = c10[v] + bb0, t11 = c11[v] + bb1;
    if (relu) {
      t00 = fmaxf(t00, 0.f); t01 = fmaxf(t01, 0.f);
      t10 = fmaxf(t10, 0.f); t11 = fmaxf(t11, 0.f);
    }
    c00[v] = t00; c01[v] = t01; c10[v] = t10; c11[v] = t11;
  }
  if (ncol0 < OC) {
    float* o = out + ((size_t)n * OC + ncol0) * HWSZ + lbase + rowb;
    *(float4*)(o)      = make_float4(c00[0], c00[1], c00[2], c00[3]);
    *(float4*)(o + 4)  = make_float4(c00[4], c00[5], c00[6], c00[7]);
    *(float4*)(o + 16) = make_float4(c10[0], c10[1], c10[2], c10[3]);
    *(float4*)(o + 20) = make_float4(c10[4], c10[5], c10[6], c10[7]);
  }
  if (ncol1 < OC) {
    float* o = out + ((size_t)n * OC + ncol1) * HWSZ + lbase + rowb;
    *(float4*)(o)      = make_float4(c01[0], c01[1], c01[2], c01[3]);
    *(float4*)(o + 4)  = make_float4(c01[4], c01[5], c01[6], c01[7]);
    *(float4*)(o + 16) = make_float4(c11[0], c11[1], c11[2], c11[3]);
    *(float4*)(o + 20) = make_float4(c11[4], c11[5], c11[6], c11[7]);
  }
}

// ---------------------------------------------------------------------------
// Deformable 3x3 conv (IC=OC=128): implicit GEMM, bilinear-gathered A shared
// across 4 N-tiles (gather cost amortized 4x). grid = (N*HW/16, 2), block=32.
// ---------------------------------------------------------------------------
__global__ __launch_bounds__(32) void k_deform(const float* __restrict__ in,
                                               const float* __restrict__ offs,
                                               const float* __restrict__ w,
                                               const float* __restrict__ bias,
                                               float* __restrict__ out) {
  int lane = threadIdx.x & 31;
  int lr = lane & 15;
  int hi = lane >> 4;
  int kofs = hi << 1;
  int mt = blockIdx.x;
  int n = mt >> 7;
  int lbase = (mt & 127) << 4;
  int l = lbase + lr;
  int y = l >> 6, x = l & 63;
  int ncol[4];
#pragma unroll
  for (int j = 0; j < 4; j++) ncol[j] = (blockIdx.y << 6) + (j << 4) + lr;
  const float* nbase = in + (size_t)n * CC * HWSZ;
  v8f acc[4] = {{}, {}, {}, {}};
  for (int kk = 0; kk < 9; kk++) {
    float dy = offs[((size_t)n * 18 + kk * 2) * HWSZ + l];
    float dx = offs[((size_t)n * 18 + kk * 2 + 1) * HWSZ + l];
    float py = (float)y + (float)(kk / 3 - 1) + dy;
    float px = (float)x + (float)(kk % 3 - 1) + dx;
    float y0f = floorf(py), x0f = floorf(px);
    int y0 = (int)y0f, x0 = (int)x0f;
    float wy = py - y0f, wx = px - x0f;
    int i0[4]; float wv[4];
#pragma unroll
    for (int cI = 0; cI < 4; cI++) {
      int yy = y0 + (cI >> 1);
      int xx = x0 + (cI & 1);
      int valid = (yy >= 0) & (yy < HH) & (xx >= 0) & (xx < WW);
      int cy = yy < 0 ? 0 : (yy >= HH ? HH - 1 : yy);
      int cx = xx < 0 ? 0 : (xx >= WW ? WW - 1 : xx);
      i0[cI] = cy * WW + cx;
      float wc = ((cI >> 1) ? wy : (1.f - wy)) * ((cI & 1) ? wx : (1.f - wx));
      wv[cI] = valid ? wc : 0.f;
    }
    const float* bp[4];
#pragma unroll
    for (int j = 0; j < 4; j++) bp[j] = w + (size_t)ncol[j] * CC * 9 + kk;
    for (int c = 0; c < CC; c += 4) {
      int cc = c + kofs;
      const float* p0 = nbase + (size_t)cc * HWSZ;
      const float* p1 = nbase + (size_t)(cc + 1) * HWSZ;
      v2f a;
      a.x = wv[0] * p0[i0[0]] + wv[1] * p0[i0[1]] + wv[2] * p0[i0[2]] +
            wv[3] * p0[i0[3]];
      a.y = wv[0] * p1[i0[0]] + wv[1] * p1[i0[1]] + wv[2] * p1[i0[2]] +
            wv[3] * p1[i0[3]];
#pragma unroll
      for (int j = 0; j < 4; j++) {
        v2f b;
        b.x = bp[j][(size_t)cc * 9];
        b.y = bp[j][(size_t)(cc + 1) * 9];
        acc[j] = wmma4(a, b, acc[j]);
      }
    }
  }
  int rowb = hi << 3;
#pragma unroll
  for (int j = 0; j < 4; j++) {
    float bb = bias[ncol[j]];
#pragma unroll
    for (int v = 0; v < 8; v++) acc[j][v] = fmaxf(acc[j][v] + bb, 0.f);
    float* o = out + ((size_t)n * CC + ncol[j]) * HWSZ + lbase + rowb;
    *(float4*)(o)     = make_float4(acc[j][0], acc[j][1], acc[j][2], acc[j][3]);
    *(float4*)(o + 4) = make_float4(acc[j][4], acc[j][5], acc[j][6], acc[j][7]);
  }
}

// ---------------------------------------------------------------------------
// Online softmax update for one 16x16 score tile in WMMA C-layout.
// Each row lives entirely in one 16-lane half-group -> shfl_xor width 16.
// ---------------------------------------------------------------------------
static __device__ inline void online_softmax(const v8f& s8, float scale,
                                             float* mrun, float* lrun,
                                             float* p, float* sc) {
#pragma unroll
  for (int v = 0; v < 8; v++) {
    float sv = s8[v] * scale;
    float mx = sv;
    mx = fmaxf(mx, __shfl_xor(mx, 1, 16));
    mx = fmaxf(mx, __shfl_xor(mx, 2, 16));
    mx = fmaxf(mx, __shfl_xor(mx, 4, 16));
    mx = fmaxf(mx, __shfl_xor(mx, 8, 16));
    float mnew = fmaxf(mrun[v], mx);
    float pv = __expf(sv - mnew);
    float rs = pv;
    rs += __shfl_xor(rs, 1, 16);
    rs += __shfl_xor(rs, 2, 16);
    rs += __shfl_xor(rs, 4, 16);
    rs += __shfl_xor(rs, 8, 16);
    float scl = __expf(mrun[v] - mnew);
    lrun[v] = lrun[v] * scl + rs;
    mrun[v] = mnew;
    sc[v] = scl;
    p[v] = pv;
  }
}

// ---------------------------------------------------------------------------
// Flash attention: 4 heads, d=32, L=2048. qkv is channel-major [n][384][L].
// One wave per 32-query block (two 16-query tiles sharing all K/V traffic).
// grid = N*4*64, block = 32.  Output att[n][e][L] (e = head*32 + d).
// ---------------------------------------------------------------------------
__global__ __launch_bounds__(32) void k_attn(const float* __restrict__ qkv,
                                             float* __restrict__ att) {
  const int L = HWSZ;
  int lane = threadIdx.x & 31;
  int lr = lane & 15;
  int hi = lane >> 4;
  int kofs = hi << 1;
  int bid = blockIdx.x;
  int qt = bid & 63;
  int head = (bid >> 6) & 3;
  int n = bid >> 8;
  int qbase = qt << 5;
  const float* qp = qkv + ((size_t)n * 384 + head * 32) * L;
  const float* kp = qp + (size_t)128 * L;
  const float* vp = qp + (size_t)256 * L;
  const float scale = 0.17677669529663687f;  // 1/sqrt(32)

  // Two Q tiles in A layout, reused over all key tiles
  v2f qa0[8], qa1[8];
  int lq0 = qbase + lr, lq1 = lq0 + 16;
#pragma unroll
  for (int s = 0; s < 8; s++) {
    int d0 = 4 * s + kofs;
    qa0[s].x = qp[(size_t)d0 * L + lq0];
    qa0[s].y = qp[(size_t)(d0 + 1) * L + lq0];
    qa1[s].x = qp[(size_t)d0 * L + lq1];
    qa1[s].y = qp[(size_t)(d0 + 1) * L + lq1];
  }

  float m0r[8], l0r[8], m1r[8], l1r[8];
  v8f o00 = {}, o01 = {}, o10 = {}, o11 = {};
#pragma unroll
  for (int v = 0; v < 8; v++) {
    m0r[v] = -1e30f; l0r[v] = 0.f;
    m1r[v] = -1e30f; l1r[v] = 0.f;
  }

  __shared__ float ldsP[2 * 256];

  for (int kt = 0; kt < 128; kt++) {
    int kb = kt << 4;
    if (kt + 1 < 128) {  // prefetch next K/V tile (global_prefetch_b8)
      __builtin_prefetch(kp + (size_t)lane * L + kb + 16, 0, 1);
      __builtin_prefetch(vp + (size_t)lane * L + kb + 16, 0, 1);
    }
    // K tile operands, shared by both query tiles
    v2f kbv[8];
#pragma unroll
    for (int s = 0; s < 8; s++) {
      int d0 = 4 * s + kofs;
      kbv[s].x = kp[(size_t)d0 * L + kb + lr];
      kbv[s].y = kp[(size_t)(d0 + 1) * L + kb + lr];
    }
    v8f s0 = {}, s1 = {};
#pragma unroll
    for (int s = 0; s < 8; s++) {
      s0 = wmma4(qa0[s], kbv[s], s0);
      s1 = wmma4(qa1[s], kbv[s], s1);
    }
    float p0[8], sc0[8], p1[8], sc1[8];
    online_softmax(s0, scale, m0r, l0r, p0, sc0);
    online_softmax(s1, scale, m1r, l1r, p1, sc1);
#pragma unroll
    for (int v = 0; v < 8; v++) {
      o00[v] *= sc0[v]; o01[v] *= sc0[v];
      o10[v] *= sc1[v]; o11[v] *= sc1[v];
    }
    // C-layout -> A-layout transpose of both P tiles through LDS
    __syncthreads();
#pragma unroll
    for (int v = 0; v < 8; v++) {
      ldsP[(v + (hi << 3)) * 16 + lr] = p0[v];
      ldsP[256 + (v + (hi << 3)) * 16 + lr] = p1[v];
    }
    __syncthreads();
    // O += P * V  (V operands shared by both query tiles)
#pragma unroll
    for (int s = 0; s < 4; s++) {
      int k0 = 4 * s + kofs;
      v2f a0, a1, b0, b1;
      a0.x = ldsP[lr * 16 + k0];
      a0.y = ldsP[lr * 16 + k0 + 1];
      a1.x = ldsP[256 + lr * 16 + k0];
      a1.y = ldsP[256 + lr * 16 + k0 + 1];
      b0.x = vp[(size_t)lr * L + kb + k0];
      b0.y = vp[(size_t)lr * L + kb + k0 + 1];
      b1.x = vp[(size_t)(lr + 16) * L + kb + k0];
      b1.y = vp[(size_t)(lr + 16) * L + kb + k0 + 1];
      o00 = wmma4(a0, b0, o00);
      o01 = wmma4(a0, b1, o01);
      o10 = wmma4(a1, b0, o10);
      o11 = wmma4(a1, b1, o11);
    }
  }
  int rowb = hi << 3;
#pragma unroll
  for (int v = 0; v < 8; v++) {
    float i0 = 1.f / l0r[v], i1 = 1.f / l1r[v];
    o00[v] *= i0; o01[v] *= i0;
    o10[v] *= i1; o11[v] *= i1;
  }
  float* a00 = att + ((size_t)n * 128 + head * 32 + lr) * L + qbase + rowb;
  float* a01 = a00 + (size_t)16 * L;
  *(float4*)(a00)      = make_float4(o00[0], o00[1], o00[2], o00[3]);
  *(float4*)(a00 + 4)  = make_float4(o00[4], o00[5], o00[6], o00[7]);
  *(float4*)(a00 + 16) = make_float4(o10[0], o10[1], o10[2], o10[3]);
  *(float4*)(a00 + 20) = make_float4(o10[4], o10[5], o10[6], o10[7]);
  *(float4*)(a01)      = make_float4(o01[0], o01[1], o01[2], o01[3]);
  *(float4*)(a01 + 4)  = make_float4(o01[4], o01[5], o01[6], o01[7]);
  *(float4*)(a01 + 16) = make_float4(o11[0], o11[1], o11[2], o11[3]);
  *(float4*)(a01 + 20) = make_float4(o11[4], o11[5], o11[6], o11[7]);
}

// ---------------------------------------------------------------------------
extern "C" void kernel_launch(void* const* d_in, const int* in_sizes, int n_in,
                              void* d_out, int out_size, void* d_ws,
                              size_t ws_size, hipStream_t stream) {
  const float* x        = (const float*)d_in[0];
  const float* conv1_w  = (const float*)d_in[1];
  const float* conv1_b  = (const float*)d_in[2];
  const float* conv2_w  = (const float*)d_in[3];
  const float* conv2_b  = (const float*)d_in[4];
  const float* off_w    = (const float*)d_in[5];
  const float* off_b    = (const float*)d_in[6];
  const float* dc_w     = (const float*)d_in[7];
  const float* dc_b     = (const float*)d_in[8];
  const float* conv3_w  = (const float*)d_in[9];
  const float* conv3_b  = (const float*)d_in[10];
  const float* conv4_w  = (const float*)d_in[11];
  const float* conv4_b  = (const float*)d_in[12];
  const float* in_w     = (const float*)d_in[13];
  const float* in_b     = (const float*)d_in[14];
  const float* out_w    = (const float*)d_in[15];
  const float* out_b    = (const float*)d_in[16];

  float* ws   = (float*)d_ws;
  float* buf1 = ws;                               // (8,256,2048)  conv1 out
  float* bufA = buf1 + (size_t)NB * 256 * HWSZ;   // (8,128,2048)
  float* bufB = bufA + (size_t)NB * 128 * HWSZ;   // (8,128,2048)
  float* offs = bufB + (size_t)NB * 128 * HWSZ;   // (8,18,2048)
  float* qkvb = offs + (size_t)NB * 18 * HWSZ;    // (8,384,2048)
  float* attb = qkvb + (size_t)NB * 384 * HWSZ;   // (8,128,2048)

  const int MT32 = NB * HWSZ / 32;  // 512 M-tiles of 32 rows
  const int MT16 = NB * HWSZ / 16;  // 1024 M-tiles of 16 rows
  dim3 blk(32);

  // conv1: 1x1, 512->256, relu
  k_gemm_cm<<<dim3(MT32, 256 / 32), blk, 0, stream>>>(x, conv1_w, conv1_b,
                                                      buf1, 512, 256, 1);
  // conv2: 3x3, 256->128, relu
  k_conv3x3<<<dim3(MT32, 128 / 32), blk, 0, stream>>>(buf1, conv2_w, conv2_b,
                                                      bufA, 256, 128, 1);
  // 4 deformable blocks
  float* hin = bufA;
  float* hout = bufB;
  for (int i = 0; i < 4; i++) {
    k_conv3x3<<<dim3(MT32, 1), blk, 0, stream>>>(
        hin, off_w + (size_t)i * 18 * 128 * 9, off_b + (size_t)i * 18, offs,
        128, 18, 0);
    k_deform<<<dim3(MT16, 2), blk, 0, stream>>>(
        hin, offs, dc_w + (size_t)i * 128 * 128 * 9, dc_b + (size_t)i * 128,
        hout);
    float* t = hin; hin = hout; hout = t;
  }
  // conv3 (relu), conv4 (no relu)
  k_conv3x3<<<dim3(MT32, 128 / 32), blk, 0, stream>>>(hin, conv3_w, conv3_b,
                                                      hout, 128, 128, 1);
  k_conv3x3<<<dim3(MT32, 128 / 32), blk, 0, stream>>>(hout, conv4_w, conv4_b,
                                                      hin, 128, 128, 0);
  // qkv projection: 128 -> 384
  k_gemm_cm<<<dim3(MT32, 384 / 32), blk, 0, stream>>>(hin, in_w, in_b, qkvb,
                                                      128, 384, 0);
  // flash attention (two 16-query tiles per wave)
  k_attn<<<dim3(NB * 4 * (HWSZ / 32)), blk, 0, stream>>>(qkvb, attb);
  // out projection -> final output (N,C,H,W) channel-major
  k_gemm_cm<<<dim3(MT32, 128 / 32), blk, 0, stream>>>(attb, out_w, out_b,
                                                      (float*)d_out, 128, 128,
                                                      0);
}